// T1DCohort_11373073400313
// MI455X (gfx1250) — compile-verified
//
#include <hip/hip_runtime.h>
#include <cstdint>
#include <cstddef>

#define BLOCK 256
#define ROWS_PER_WAVE 32
#define WAVES_PER_BLOCK (BLOCK / 32)

// Low 32 bits of a generic pointer into LDS == byte offset within the
// workgroup's LDS allocation.
__device__ __forceinline__ uint32_t lds_off_u32(const void* p) {
    return (uint32_t)(uintptr_t)p;
}

__device__ __forceinline__ void async_load16(uint32_t lds_byte, const void* gaddr) {
    asm volatile("global_load_async_to_lds_b128 %0, %1, off"
                 :: "v"(lds_byte), "v"(gaddr) : "memory");
}

__device__ __forceinline__ void async_store16(void* gaddr, uint32_t lds_byte) {
    asm volatile("global_store_async_from_lds_b128 %0, %1, off"
                 :: "v"(gaddr), "v"(lds_byte) : "memory");
}

__device__ __forceinline__ void wait_async0() {
    asm volatile("s_wait_asynccnt 0x0" ::: "memory");
}

__device__ __forceinline__ void wait_ds0() {
    asm volatile("s_wait_dscnt 0x0" ::: "memory");
}

// CDNA5 native transcendental (new VOP1 on gfx1250): 1 instr instead of the
// ~30-instr ocml tanhf expansion.
__device__ __forceinline__ float fast_tanh(float a) {
    float r;
    asm("v_tanh_f32 %0, %1" : "=v"(r) : "v"(a));
    return r;
}

// ~1-ulp hardware reciprocal; kernel is HBM-bound so we trade exact-rounded
// division for a single VALU op.
__device__ __forceinline__ float fast_rcp(float a) {
    return __builtin_amdgcn_rcpf(a);
}

__device__ __forceinline__ void t1d_compute(
    const float* __restrict__ xs, const float* __restrict__ p,
    float ains, float acho, float lastQ, float lastF, float* __restrict__ dx)
{
    const float BW = p[0],  KMAX = p[1],  B   = p[2],  D   = p[3],  KMIN = p[4];
    const float KABS = p[5], F   = p[6],  KP1 = p[7],  KP2 = p[8],  KP3  = p[9];
    const float FSNC = p[10], KE1 = p[11], KE2 = p[12], K1 = p[13], K2   = p[14];
    const float VM0 = p[15], VMX = p[16], KM0 = p[17], M1 = p[18], M2   = p[19];
    const float M4  = p[20], KA1 = p[21], KA2 = p[22], VI = p[23], P2U  = p[24];
    const float IB  = p[25], KI  = p[26], M30 = p[27], KD = p[28], KSC  = p[29];

    const float x0 = xs[0], x1 = xs[1], x2 = xs[2], x3 = xs[3], x4 = xs[4];
    const float x5 = xs[5], x6 = xs[6], x7 = xs[7], x8 = xs[8], x9 = xs[9];
    const float x10 = xs[10], x11 = xs[11], x12 = xs[12];

    const float rcpBW = fast_rcp(BW);
    const float f_ins = ains * 6000.0f * rcpBW;
    const float qsto  = x0 + x1;
    const float Dbar  = lastQ + lastF;

    dx[0] = -KMAX * x0 + acho * 1000.0f;

    const float invDb = fast_rcp(Dbar + 1e-7f);
    const float aa = 2.5f * fast_rcp(1.0f - B) * invDb;
    const float cc = 2.5f * fast_rcp(D) * invDb;
    float kgut = KMIN + (KMAX - KMIN) * 0.5f *
                 (fast_tanh(aa * (qsto - B * Dbar)) -
                  fast_tanh(cc * (qsto - D * Dbar)) + 2.0f);
    kgut = (Dbar > 0.0f) ? kgut : KMAX;

    dx[1] = KMAX * x0 - x1 * kgut;
    dx[2] = kgut * x1 - KABS * x2;

    const float Rat  = F * KABS * x2 * rcpBW;
    const float EGPt = KP1 - KP2 * x3 - KP3 * x8;
    const float Et   = (x3 > KE2) ? KE1 * (x3 - KE2) : 0.0f;
    const float d3   = fmaxf(EGPt, 0.0f) + Rat - FSNC - Et - K1 * x3 + K2 * x4;
    dx[3] = (x3 >= 0.0f) ? d3 : 0.0f;

    const float Vmt  = VM0 + VMX * x6;
    const float Uidt = Vmt * x4 * fast_rcp(KM0 + x4);
    const float d4   = -Uidt + K1 * x3 - K2 * x4;
    dx[4] = (x4 >= 0.0f) ? d4 : 0.0f;

    const float d5 = -(M2 + M4) * x5 + M1 * x9 + KA1 * x10 + KA2 * x11;
    const float It = x5 * fast_rcp(VI);
    dx[5] = (x5 >= 0.0f) ? d5 : 0.0f;

    dx[6] = -P2U * x6 + P2U * (It - IB);
    dx[7] = -KI * (x7 - It);
    dx[8] = -KI * (x8 - x7);

    const float d9 = -(M1 + M30) * x9 + M2 * x5;
    dx[9] = (x9 >= 0.0f) ? d9 : 0.0f;

    const float d10 = f_ins - (KA1 + KD) * x10;
    dx[10] = (x10 >= 0.0f) ? d10 : 0.0f;

    const float d11 = KD * x10 - KA2 * x11;
    dx[11] = (x11 >= 0.0f) ? d11 : 0.0f;

    const float d12 = -KSC * x12 + KSC * x3;
    dx[12] = (x12 >= 0.0f) ? d12 : 0.0f;
}

__global__ __launch_bounds__(BLOCK) void t1d_dxdt_kernel(
    const float* __restrict__ x,     // (N,13)
    const float* __restrict__ par,   // (N,30)
    const float* __restrict__ ains,  // (N,1)
    const float* __restrict__ acho,  // (N,1)
    const float* __restrict__ lq,    // (N,1)
    const float* __restrict__ lf,    // (N,1)
    float* __restrict__ out,         // (N,13)
    int n)
{
    __shared__ __align__(16) float sx[WAVES_PER_BLOCK][ROWS_PER_WAVE * 13]; // 1664 B/wave
    __shared__ __align__(16) float sp[WAVES_PER_BLOCK][ROWS_PER_WAVE * 30]; // 3840 B/wave

    const int tid  = threadIdx.x;
    const int wave = tid >> 5;
    const int lane = tid & 31;
    const int base_row = blockIdx.x * BLOCK + wave * ROWS_PER_WAVE;
    const int row = base_row + lane;

    if (base_row + ROWS_PER_WAVE <= n) {
        // ---- Stage x tile: 32*13 floats = 104 x 16B chunks (coalesced) ----
        {
            const char* gsrc = (const char*)(x + (size_t)base_row * 13);
            const uint32_t lb = lds_off_u32(&sx[wave][0]);
            #pragma unroll
            for (int i = 0; i < 3; ++i)
                async_load16(lb + (lane + 32 * i) * 16, gsrc + (size_t)(lane + 32 * i) * 16);
            if (lane < 8)
                async_load16(lb + (lane + 96) * 16, gsrc + (size_t)(lane + 96) * 16);
        }
        // ---- Stage params tile: 32*30 floats = 240 x 16B chunks ----
        {
            const char* gsrc = (const char*)(par + (size_t)base_row * 30);
            const uint32_t lb = lds_off_u32(&sp[wave][0]);
            #pragma unroll
            for (int i = 0; i < 7; ++i)
                async_load16(lb + (lane + 32 * i) * 16, gsrc + (size_t)(lane + 32 * i) * 16);
            if (lane < 16)
                async_load16(lb + (lane + 224) * 16, gsrc + (size_t)(lane + 224) * 16);
        }

        // Already-coalesced per-row scalars: plain loads, overlap with DMA.
        const float ai = ains[row], ac = acho[row], ql = lq[row], fl = lf[row];

        wait_async0();  // own-wave async loads landed in LDS

        // Conflict-free LDS reads: dword addr 13*lane (resp. 30*lane) mod 64
        // is a permutation over the 32 lanes.
        const float* xr = &sx[wave][lane * 13];
        const float* pr = &sp[wave][lane * 30];

        float dx[13];
        t1d_compute(xr, pr, ai, ac, ql, fl, dx);

        // ---- Results -> LDS (reuse x tile region), then async store out ----
        float* wr = &sx[wave][lane * 13];
        #pragma unroll
        for (int k = 0; k < 13; ++k) wr[k] = dx[k];

        wait_ds0();  // commit DS stores before the DMA engine reads LDS

        {
            char* gdst = (char*)(out + (size_t)base_row * 13);
            const uint32_t lb = lds_off_u32(&sx[wave][0]);
            #pragma unroll
            for (int i = 0; i < 3; ++i)
                async_store16(gdst + (size_t)(lane + 32 * i) * 16, lb + (lane + 32 * i) * 16);
            if (lane < 8)
                async_store16(gdst + (size_t)(lane + 96) * 16, lb + (lane + 96) * 16);
        }
        wait_async0();
    } else if (row < n) {
        // Scalar tail path (not taken for N = 2^20, kept for generality).
        float xr[13], pr[30], dx[13];
        #pragma unroll
        for (int k = 0; k < 13; ++k) xr[k] = x[(size_t)row * 13 + k];
        #pragma unroll
        for (int k = 0; k < 30; ++k) pr[k] = par[(size_t)row * 30 + k];
        t1d_compute(xr, pr, ains[row], acho[row], lq[row], lf[row], dx);
        #pragma unroll
        for (int k = 0; k < 13; ++k) out[(size_t)row * 13 + k] = dx[k];
    }
}

extern "C" void kernel_launch(void* const* d_in, const int* in_sizes, int n_in,
                              void* d_out, int out_size, void* d_ws, size_t ws_size,
                              hipStream_t stream) {
    (void)n_in; (void)d_ws; (void)ws_size; (void)out_size;
    const float* x    = (const float*)d_in[0];
    const float* par  = (const float*)d_in[1];
    const float* ains = (const float*)d_in[2];
    const float* acho = (const float*)d_in[3];
    const float* lq   = (const float*)d_in[4];
    const float* lf   = (const float*)d_in[5];
    float* out        = (float*)d_out;

    const int n = in_sizes[2];  // action_ins has N elements
    const int grid = (n + BLOCK - 1) / BLOCK;
    t1d_dxdt_kernel<<<grid, BLOCK, 0, stream>>>(x, par, ains, acho, lq, lf, out, n);
}